// SemanticPairLoss_80298708566624
// MI455X (gfx1250) — compile-verified
//
#include <hip/hip_runtime.h>
#include <hip/hip_bf16.h>

typedef __attribute__((ext_vector_type(2))) float v2f;
typedef __attribute__((ext_vector_type(8))) float v8f;

#define ALPHA_C 0.5f
#define IMG 3072          // 3*32*32
#define NB 16
// ps = 8 stage
#define N8 625
#define N8P 640
#define M8 25
#define NC8 3125
#define NC8P 3136         // 196 tiles of 16 (even)
#define D8 192
// ps = 4 stage
#define N4 841
#define N4P 848
#define M4 29
#define NC4 4205
#define NC4P 4224         // 264 tiles of 16 (even)
#define D4 48

__device__ __forceinline__ float px(const float* img, int c, int y, int x) {
    return img[c * 1024 + y * 32 + x];
}

// ---------------------------------------------------------------------------
// Kernel 1: build pair-interleaved operands.
//   PA[(K/2)*NP  + row] = {Ahat[K][row], Ahat[K+1][row]}   (K even)
//   PB[(K/2)*NCP + col] = {B[K][col],    B[K+1][col]}
// Each WMMA fragment is then a single contiguous global_load_b64 per lane.
// ---------------------------------------------------------------------------
__global__ void build_kernel(const float* __restrict__ inp, const float* __restrict__ gt,
                             const float* __restrict__ cands,
                             float2* __restrict__ PA8, float* __restrict__ RC8,
                             float2* __restrict__ PB8, float* __restrict__ BB8,
                             float2* __restrict__ PA4, float* __restrict__ RC4,
                             float2* __restrict__ PB4, float* __restrict__ BB4)
{
    const int b = blockIdx.y;
    const int t = blockIdx.x * blockDim.x + threadIdx.x;
    const float* gi = gt + b * IMG;
    const float* pi = inp + b * IMG;
    const float* ci = cands + (size_t)b * 4 * IMG;

    // ---- ps=8 blended A rows + row const ----
    if (t < N8P) {
        float2* A = PA8 + (size_t)b * (D8 / 2) * N8P;
        if (t < N8) {
            int i0 = t / M8, j0 = t % M8;
            float rc = 0.f, even = 0.f; int d = 0;
            for (int c = 0; c < 3; c++)
                for (int dy = 0; dy < 8; dy++)
                    for (int dx = 0; dx < 8; dx++, d++) {
                        float g = px(gi, c, i0 + dy, j0 + dx);
                        float p = px(pi, c, i0 + dy, j0 + dx);
                        float v = ALPHA_C * g + (1.f - ALPHA_C) * p;
                        rc += ALPHA_C * g * g + (1.f - ALPHA_C) * p * p;
                        if ((d & 1) == 0) even = v;
                        else A[(size_t)(d >> 1) * N8P + t] = make_float2(even, v);
                    }
            RC8[b * N8P + t] = rc;
        } else {
            for (int d = 0; d < D8 / 2; d++) A[(size_t)d * N8P + t] = make_float2(0.f, 0.f);
            RC8[b * N8P + t] = 0.f;
        }
    }
    // ---- ps=8 candidate B columns + norms ----
    if (t < NC8P) {
        float2* Bm = PB8 + (size_t)b * (D8 / 2) * NC8P;
        if (t < NC8) {
            const float* src; int p;
            if (t < 4 * N8) { src = ci + (t / N8) * IMG; p = t % N8; }
            else            { src = gi;                  p = t - 4 * N8; }
            int i0 = p / M8, j0 = p % M8;
            float bbv = 0.f, even = 0.f; int d = 0;
            for (int c = 0; c < 3; c++)
                for (int dy = 0; dy < 8; dy++)
                    for (int dx = 0; dx < 8; dx++, d++) {
                        float v = px(src, c, i0 + dy, j0 + dx);
                        bbv += v * v;
                        if ((d & 1) == 0) even = v;
                        else Bm[(size_t)(d >> 1) * NC8P + t] = make_float2(even, v);
                    }
            BB8[b * NC8P + t] = bbv;
        } else {
            for (int d = 0; d < D8 / 2; d++) Bm[(size_t)d * NC8P + t] = make_float2(0.f, 0.f);
            BB8[b * NC8P + t] = 1e30f;   // padded columns never win argmin
        }
    }
    // ---- ps=4 blended A rows ----
    if (t < N4P) {
        float2* A = PA4 + (size_t)b * (D4 / 2) * N4P;
        if (t < N4) {
            int i0 = t / M4, j0 = t % M4;
            float rc = 0.f, even = 0.f; int d = 0;
            for (int c = 0; c < 3; c++)
                for (int dy = 0; dy < 4; dy++)
                    for (int dx = 0; dx < 4; dx++, d++) {
                        float g = px(gi, c, i0 + dy, j0 + dx);
                        float p = px(pi, c, i0 + dy, j0 + dx);
                        float v = ALPHA_C * g + (1.f - ALPHA_C) * p;
                        rc += ALPHA_C * g * g + (1.f - ALPHA_C) * p * p;
                        if ((d & 1) == 0) even = v;
                        else A[(size_t)(d >> 1) * N4P + t] = make_float2(even, v);
                    }
            RC4[b * N4P + t] = rc;
        } else {
            for (int d = 0; d < D4 / 2; d++) A[(size_t)d * N4P + t] = make_float2(0.f, 0.f);
            RC4[b * N4P + t] = 0.f;
        }
    }
    // ---- ps=4 candidate B columns ----
    if (t < NC4P) {
        float2* Bm = PB4 + (size_t)b * (D4 / 2) * NC4P;
        if (t < NC4) {
            const float* src; int p;
            if (t < 4 * N4) { src = ci + (t / N4) * IMG; p = t % N4; }
            else            { src = gi;                  p = t - 4 * N4; }
            int i0 = p / M4, j0 = p % M4;
            float bbv = 0.f, even = 0.f; int d = 0;
            for (int c = 0; c < 3; c++)
                for (int dy = 0; dy < 4; dy++)
                    for (int dx = 0; dx < 4; dx++, d++) {
                        float v = px(src, c, i0 + dy, j0 + dx);
                        bbv += v * v;
                        if ((d & 1) == 0) even = v;
                        else Bm[(size_t)(d >> 1) * NC4P + t] = make_float2(even, v);
                    }
            BB4[b * NC4P + t] = bbv;
        } else {
            for (int d = 0; d < D4 / 2; d++) Bm[(size_t)d * NC4P + t] = make_float2(0.f, 0.f);
            BB4[b * NC4P + t] = 1e30f;
        }
    }
}

// ---------------------------------------------------------------------------
// Kernel 2/3: fused GEMM (f32 WMMA 16x16x4, 2-wide column blocking) +
// per-row argmin of loss(i,j)*D = rc[i] + bb[j] - 2 * (Ahat @ B^T)(i,j)
// A tile (16 x D) is preloaded once per row-tile into registers and reused
// across all column tiles; __launch_bounds__(256,1) releases the VGPR cap so
// the D=192 instantiation (48 v2f = 96 VGPRs) does not spill.
// Fragment layout (ISA 7.12.2, 32-bit 16x4): lanes<16 hold K={4k,4k+1},
// lanes>=16 hold K={4k+2,4k+3} -> one float2 load at pair index 2k+hi.
// C/D: VGPR r -> M=r (lanes 0-15) / M=r+8 (lanes 16-31), N = lane&15.
// ---------------------------------------------------------------------------
template<int D, int NP, int NCP>
__global__ void __launch_bounds__(256, 1)
wmma_argmin_kernel(const float2* __restrict__ PA, const float2* __restrict__ PB,
                   const float* __restrict__ rc, const float* __restrict__ bb,
                   float* __restrict__ ml, int* __restrict__ sel, int Nrows)
{
    const int b = blockIdx.y;
    const int row0 = blockIdx.x * 16;
    PA += (size_t)b * (D / 2) * NP;
    PB += (size_t)b * (D / 2) * NCP;
    rc += b * NP; bb += b * NCP; ml += b * NP; sel += b * NP;

    const int lane = threadIdx.x & 31;
    const int w    = threadIdx.x >> 5;      // 8 waves / block
    const int hi   = lane >> 4;
    const int l15  = lane & 15;

    // ---- preload A fragments for this 16-row tile (kept in VGPRs) ----
    const float2* pa = PA + (size_t)hi * NP + row0 + l15;
    v2f afr[D / 4];
    #pragma unroll
    for (int k = 0; k < D / 4; k++) {
        float2 af = pa[2 * k * NP];
        afr[k][0] = af.x; afr[k][1] = af.y;
    }

    float rcv[8];
    #pragma unroll
    for (int r = 0; r < 8; r++) rcv[r] = rc[row0 + r + hi * 8];

    float best[8]; int bidx[8];
    #pragma unroll
    for (int r = 0; r < 8; r++) { best[r] = 3.4e38f; bidx[r] = 0x7fffffff; }

    const int NT2 = NCP / 32;               // 2-wide tile groups
    for (int t2 = w; t2 < NT2; t2 += 8) {
        const int col0 = t2 * 32;
        const float2* pb = PB + (size_t)hi * NCP + col0 + l15;
        v8f c0 = {0.f, 0.f, 0.f, 0.f, 0.f, 0.f, 0.f, 0.f};
        v8f c1 = {0.f, 0.f, 0.f, 0.f, 0.f, 0.f, 0.f, 0.f};
        // double-buffered B fragments: overlap next step's loads with WMMAs
        float2 b0 = pb[0];
        float2 b1 = pb[16];
        #pragma unroll
        for (int k = 0; k < D / 4; k++) {
            float2 nb0, nb1;
            if (k + 1 < D / 4) {
                nb0 = pb[2 * (k + 1) * NCP];
                nb1 = pb[2 * (k + 1) * NCP + 16];
            }
            v2f bv0 = {b0.x, b0.y};
            v2f bv1 = {b1.x, b1.y};
            c0 = __builtin_amdgcn_wmma_f32_16x16x4_f32(false, afr[k], false, bv0,
                                                       (short)0, c0, false, false);
            c1 = __builtin_amdgcn_wmma_f32_16x16x4_f32(false, afr[k], false, bv1,
                                                       (short)0, c1, false, false);
            b0 = nb0; b1 = nb1;
        }
        const float bb0 = bb[col0 + l15];
        const float bb1 = bb[col0 + 16 + l15];
        const int   n0  = col0 + l15;
        const int   n1  = n0 + 16;
        #pragma unroll
        for (int r = 0; r < 8; r++) {
            float l0 = rcv[r] + bb0 - 2.f * c0[r];
            if (l0 < best[r] || (l0 == best[r] && n0 < bidx[r])) { best[r] = l0; bidx[r] = n0; }
            float l1 = rcv[r] + bb1 - 2.f * c1[r];
            if (l1 < best[r] || (l1 == best[r] && n1 < bidx[r])) { best[r] = l1; bidx[r] = n1; }
        }
    }
    // min-reduce across the 16 lanes of each half-wave (stable tie-break)
    #pragma unroll
    for (int r = 0; r < 8; r++) {
        #pragma unroll
        for (int off = 8; off >= 1; off >>= 1) {
            float ov = __shfl_xor(best[r], off, 32);
            int   oi = __shfl_xor(bidx[r], off, 32);
            if (ov < best[r] || (ov == best[r] && oi < bidx[r])) { best[r] = ov; bidx[r] = oi; }
        }
    }
    __shared__ float smin[8][16];
    __shared__ int   sidx[8][16];
    if (l15 == 0) {
        #pragma unroll
        for (int r = 0; r < 8; r++) { smin[w][r + hi * 8] = best[r]; sidx[w][r + hi * 8] = bidx[r]; }
    }
    __syncthreads();
    if (threadIdx.x < 16) {
        float bv = smin[0][threadIdx.x]; int bi = sidx[0][threadIdx.x];
        for (int ww = 1; ww < 8; ww++) {
            float ov = smin[ww][threadIdx.x]; int oi = sidx[ww][threadIdx.x];
            if (ov < bv || (ov == bv && oi < bi)) { bv = ov; bi = oi; }
        }
        int row = row0 + threadIdx.x;
        if (row < Nrows) { ml[row] = bv / (float)D; sel[row] = bi; }
    }
}

// ---------------------------------------------------------------------------
// Kernel 4/5: stable rank (argsort[:N/2] membership) + LDS scatter-average
// ---------------------------------------------------------------------------
__global__ void rank_scatter_kernel(const float* __restrict__ ml, const int* __restrict__ sel,
                                    const float* __restrict__ cands, const float* __restrict__ gtimg,
                                    const float* __restrict__ base, float* __restrict__ outreg,
                                    int N, int NP, int mdim, int ps, int perimg)
{
    const int b = blockIdx.x;
    const int t = threadIdx.x;
    __shared__ float sml[N4];       // max N = 841
    __shared__ float acc[IMG];
    __shared__ float cnt[IMG];
    ml += b * NP; sel += b * NP;
    const float* gi = gtimg + b * IMG;
    const float* ci = cands + (size_t)b * 4 * IMG;
    const float* bs = base + b * IMG;

    for (int i = t; i < N; i += blockDim.x) sml[i] = ml[i];
    for (int e = t; e < IMG; e += blockDim.x) { acc[e] = 0.f; cnt[e] = 0.f; }
    __syncthreads();

    const int nvalid = N / 2;
    for (int i = t; i < N; i += blockDim.x) {
        float v = sml[i];
        int rank = 0;
        for (int k = 0; k < N; k++) {
            float u = sml[k];
            rank += (u < v) || (u == v && k < i);   // stable argsort membership
        }
        if (rank < nvalid) {
            int j = sel[i];
            const float* src; int p;
            if (j < 4 * perimg) { src = ci + (j / perimg) * IMG; p = j % perimg; }
            else                { src = gi;                      p = j - 4 * perimg; }
            int ci0 = p / mdim, cj0 = p % mdim;
            int pi0 = i / mdim, pj0 = i % mdim;
            for (int c = 0; c < 3; c++)
                for (int dy = 0; dy < ps; dy++)
                    for (int dx = 0; dx < ps; dx++) {
                        float val = src[c * 1024 + (ci0 + dy) * 32 + (cj0 + dx)];
                        int   o   = c * 1024 + (pi0 + dy) * 32 + (pj0 + dx);
                        atomicAdd(&acc[o], val);
                        atomicAdd(&cnt[o], 1.f);
                    }
        }
    }
    __syncthreads();
    for (int e = t; e < IMG; e += blockDim.x)
        outreg[(size_t)b * IMG + e] = (acc[e] + bs[e]) / (cnt[e] + 1.f);
}

// ---------------------------------------------------------------------------
// Kernel 6: mean |inp - paired| -> scalar (single block, deterministic)
// ---------------------------------------------------------------------------
__global__ void loss_kernel(const float* __restrict__ inp, const float* __restrict__ paired,
                            float* __restrict__ out)
{
    const int t = threadIdx.x;
    float s = 0.f;
    for (int i = t; i < NB * IMG; i += blockDim.x) s += fabsf(inp[i] - paired[i]);
    #pragma unroll
    for (int off = 16; off >= 1; off >>= 1) s += __shfl_xor(s, off, 32);
    __shared__ float wsum[8];
    if ((t & 31) == 0) wsum[t >> 5] = s;
    __syncthreads();
    if (t == 0) {
        float tot = 0.f;
        for (int i = 0; i < 8; i++) tot += wsum[i];
        out[0] = tot / (float)(NB * IMG);
    }
}

// ---------------------------------------------------------------------------
extern "C" void kernel_launch(void* const* d_in, const int* in_sizes, int n_in,
                              void* d_out, int out_size, void* d_ws, size_t ws_size,
                              hipStream_t stream)
{
    (void)in_sizes; (void)n_in; (void)out_size; (void)ws_size;
    const float* inp   = (const float*)d_in[0];
    const float* gt    = (const float*)d_in[1];
    const float* cands = (const float*)d_in[2];

    float* ws = (float*)d_ws;
    size_t off = 0;
    float2* PA8 = (float2*)(ws + off); off += (size_t)NB * (D8 / 2) * N8P * 2;
    float2* PB8 = (float2*)(ws + off); off += (size_t)NB * (D8 / 2) * NC8P * 2;
    float*  RC8 = ws + off; off += (size_t)NB * N8P;
    float*  BB8 = ws + off; off += (size_t)NB * NC8P;
    float*  ML8 = ws + off; off += (size_t)NB * N8P;
    int*    SEL8 = (int*)(ws + off); off += (size_t)NB * N8P;
    float2* PA4 = (float2*)(ws + off); off += (size_t)NB * (D4 / 2) * N4P * 2;
    float2* PB4 = (float2*)(ws + off); off += (size_t)NB * (D4 / 2) * NC4P * 2;
    float*  RC4 = ws + off; off += (size_t)NB * N4P;
    float*  BB4 = ws + off; off += (size_t)NB * NC4P;
    float*  ML4 = ws + off; off += (size_t)NB * N4P;
    int*    SEL4 = (int*)(ws + off); off += (size_t)NB * N4P;
    float*  REG1   = ws + off; off += (size_t)NB * IMG;
    float*  PAIRED = ws + off; off += (size_t)NB * IMG;

    build_kernel<<<dim3(32, NB), 256, 0, stream>>>(inp, gt, cands,
                                                   PA8, RC8, PB8, BB8,
                                                   PA4, RC4, PB4, BB4);
    wmma_argmin_kernel<D8, N8P, NC8P><<<dim3(N8P / 16, NB), 256, 0, stream>>>(
        PA8, PB8, RC8, BB8, ML8, SEL8, N8);
    wmma_argmin_kernel<D4, N4P, NC4P><<<dim3(N4P / 16, NB), 256, 0, stream>>>(
        PA4, PB4, RC4, BB4, ML4, SEL4, N4);
    // stage 1: base region = gt
    rank_scatter_kernel<<<NB, 256, 0, stream>>>(ML8, SEL8, cands, gt, gt, REG1,
                                                N8, N8P, M8, 8, N8);
    // stage 2: base region = stage-1 output
    rank_scatter_kernel<<<NB, 256, 0, stream>>>(ML4, SEL4, cands, gt, REG1, PAIRED,
                                                N4, N4P, M4, 4, N4);
    loss_kernel<<<1, 256, 0, stream>>>(inp, PAIRED, (float*)d_out);
}